// SimpleSSM_26242250178952
// MI455X (gfx1250) — compile-verified
//
#include <hip/hip_runtime.h>

namespace {
constexpr int BATCH  = 4;
constexpr int T      = 4096;
constexpr int D      = 2048;
constexpr int L      = 256;            // chunk length (power of two: A^L via squaring)
constexpr int NC     = T / L;          // 16 chunks
constexpr int SUB    = 32;             // timesteps per LDS sub-tile
constexpr int NSUB   = L / SUB;        // 8 sub-tiles per chunk
constexpr int DT     = 32;             // channels per wave (one lane per channel)
constexpr int DTILES = D / DT;         // 64 d-tiles
constexpr int WPB    = 8;              // waves per block
constexpr int NBUF   = 3;              // triple-buffered async pipeline
}

// Issue one sub-tile (SUB rows x 128B) as 8 async b128 loads.
// Each instruction: 32 lanes x 16B -> 4 rows of 128B. Tracked by ASYNCcnt.
__device__ __forceinline__ void async_tile_load(unsigned lds_addr,
                                                unsigned long long gaddr)
{
#pragma unroll
    for (int j = 0; j < 8; ++j) {
        asm volatile("global_load_async_to_lds_b128 %0, %1, off"
                     :: "v"(lds_addr), "v"(gaddr)
                     : "memory");
        lds_addr += 4u * DT * 4u;                       // +512 B in LDS
        gaddr    += 4ull * (unsigned long long)D * 4u;  // +32768 B in memory
    }
}

// PASS 0: compute per-chunk carry (h starting from 0)
// PASS 1: recompute chunk with correct initial state, emit y
template <int PASS>
__launch_bounds__(WPB * 32)
__global__ void ssm_scan_kernel(const float* __restrict__ x,
                                const float* __restrict__ Av,
                                const float* __restrict__ Bv,
                                const float* __restrict__ Cv,
                                const float* __restrict__ Dv,
                                float* __restrict__ y,
                                float* __restrict__ carry,
                                const float* __restrict__ hinit)
{
    __shared__ float lds[WPB][NBUF][SUB * DT];   // 8 waves * 3 buffers * 4KB = 96KB

    const int lane  = threadIdx.x & 31;
    const int wib   = threadIdx.x >> 5;
    const int W     = blockIdx.x * WPB + wib;       // global wave id
    const int dtile = W % DTILES;
    const int chunk = (W / DTILES) % NC;
    const int b     = W / (DTILES * NC);
    const int d     = dtile * DT + lane;

    // Per-lane base for async tile loads: lane covers row (lane>>3) of each
    // 4-row group, 16-byte column (lane&7).
    const unsigned long long gbase =
        (unsigned long long)(const char*)(x + (size_t)b * T * D +
                                          (size_t)chunk * L * D +
                                          (size_t)dtile * DT) +
        (unsigned long long)(lane >> 3) * ((unsigned long long)D * 4u) +
        (unsigned long long)(lane & 7) * 16u;

    const unsigned laneoff = (unsigned)((lane >> 3) * (DT * 4) + (lane & 7) * 16);
    unsigned ldsbase[NBUF];
#pragma unroll
    for (int k = 0; k < NBUF; ++k)
        ldsbase[k] = (unsigned)(uintptr_t)&lds[wib][k][0] + laneoff;

    constexpr unsigned long long SUBBYTES = (unsigned long long)SUB * D * 4u;

    const float a  = Av[d];
    const float bb = Bv[d];
    float c = 0.f, dd = 0.f;
    float s = 0.f;
    float* yp = nullptr;
    if (PASS == 1) {
        c  = Cv[d];
        dd = Dv[d];
        s  = hinit[((size_t)b * NC + chunk) * D + d];
        yp = y + (size_t)b * T * D + (size_t)chunk * L * D + d;
    }

    // Prime the pipeline two sub-tiles deep.
    async_tile_load(ldsbase[0], gbase);
    async_tile_load(ldsbase[1], gbase + SUBBYTES);

    for (int st = 0; st < NSUB; ++st) {
        if (st + 2 < NSUB) {
            // Recycled buffer: make sure our previous sub-tile's DS reads are
            // fully retired before the async engine overwrites it (async LDS
            // writes are NOT ordered against same-wave ds_loads).
            asm volatile("s_wait_dscnt 0" ::: "memory");
            async_tile_load(ldsbase[(st + 2) % NBUF],
                            gbase + (unsigned long long)(st + 2) * SUBBYTES);
            asm volatile("s_wait_asynccnt 16" ::: "memory");  // groups st+1,st+2 may fly
        } else if (st + 1 < NSUB) {
            asm volatile("s_wait_asynccnt 8" ::: "memory");   // group st+1 may fly
        } else {
            asm volatile("s_wait_asynccnt 0" ::: "memory");
        }

        const float* bp = &lds[wib][st % NBUF][0];
#pragma unroll
        for (int r = 0; r < SUB; ++r) {
            const float xv = bp[r * DT + lane];     // ds_load_b32, conflict-free
            s = fmaf(a, s, bb * xv);                // h = A*h + B*x
            if (PASS == 1) {
                const float yv = fmaf(c, s, dd * xv);   // y = C*h + D*x
                // Non-temporal: keep the 134MB x stream resident in the 192MB
                // L2 so the pass-3 re-read of x is served from L2, not HBM.
                __builtin_nontemporal_store(yv, yp + (size_t)(st * SUB + r) * D);
            }
        }
    }

    if (PASS == 0)
        carry[((size_t)b * NC + chunk) * D + d] = s;
}

// Sequential combine across chunks (exclusive prefix of initial states).
__global__ void ssm_combine_kernel(const float* __restrict__ Av,
                                   const float* __restrict__ carry,
                                   float* __restrict__ hinit)
{
    const int idx = blockIdx.x * blockDim.x + threadIdx.x;
    if (idx >= BATCH * D) return;
    const int b = idx / D;
    const int d = idx % D;

    float aL = Av[d];
#pragma unroll
    for (int i = 0; i < 8; ++i) aL *= aL;   // a^(2^8) = a^256 = a^L

    float h = 0.f;
#pragma unroll
    for (int ch = 0; ch < NC; ++ch) {
        const size_t o = ((size_t)b * NC + ch) * D + d;
        hinit[o] = h;                       // state at chunk start
        h = fmaf(aL, h, carry[o]);
    }
}

extern "C" void kernel_launch(void* const* d_in, const int* in_sizes, int n_in,
                              void* d_out, int out_size, void* d_ws, size_t ws_size,
                              hipStream_t stream)
{
    const float* x  = (const float*)d_in[0];
    const float* A  = (const float*)d_in[1];
    const float* B  = (const float*)d_in[2];
    const float* C  = (const float*)d_in[3];
    const float* Dv = (const float*)d_in[4];
    float* y = (float*)d_out;

    float* carry = (float*)d_ws;                          // BATCH*NC*D floats (512KB)
    float* hinit = carry + (size_t)BATCH * NC * D;        // BATCH*NC*D floats (512KB)

    const dim3 blk(WPB * 32);
    const dim3 grid1((BATCH * NC * DTILES) / WPB);        // 512 blocks, 4096 waves

    ssm_scan_kernel<0><<<grid1, blk, 0, stream>>>(x, A, B, C, Dv,
                                                  nullptr, carry, nullptr);
    ssm_combine_kernel<<<(BATCH * D + 255) / 256, 256, 0, stream>>>(A, carry, hinit);
    ssm_scan_kernel<1><<<grid1, blk, 0, stream>>>(x, A, B, C, Dv,
                                                  y, carry, hinit);
}